// SetAbstraction_41351945126433
// MI455X (gfx1250) — compile-verified
//
#include <hip/hip_runtime.h>

// ---------------- problem constants ----------------
#define Bsz   4
#define Npts  16384
#define Mq    2048
#define Cx    32
#define Kn    32
#define CIN0  35
#define C0    64
#define C1    64
#define C2    128
#define Jtot  (Bsz * Mq * Kn)      // 262144 positions
#define MK    (Mq * Kn)            // 65536
#define R2    0.25f
#define EPSBN 1e-5f

typedef __attribute__((ext_vector_type(2))) float v2f;
typedef __attribute__((ext_vector_type(8))) float v8f;

// fp32 WMMA: D(16x16) = A(16x4) x B(4x16) + C   -> v_wmma_f32_16x16x4_f32
__device__ inline v8f wmma4(float a0, float a1, float b0, float b1, v8f c) {
  v2f a = {a0, a1};
  v2f b = {b0, b1};
  return __builtin_amdgcn_wmma_f32_16x16x4_f32(false, a, false, b, (short)0, c,
                                               false, false);
}

__device__ inline float xsum16(float v) {
  v += __shfl_xor(v, 1, 32);
  v += __shfl_xor(v, 2, 32);
  v += __shfl_xor(v, 4, 32);
  v += __shfl_xor(v, 8, 32);
  return v;
}
__device__ inline float xmax16(float v) {
  v = fmaxf(v, __shfl_xor(v, 1, 32));
  v = fmaxf(v, __shfl_xor(v, 2, 32));
  v = fmaxf(v, __shfl_xor(v, 4, 32));
  v = fmaxf(v, __shfl_xor(v, 8, 32));
  return v;
}

// ---------------- init: copy q to d_out head, zero stat sums ----------------
__global__ void init_kernel(const float* __restrict__ q, float* __restrict__ dout,
                            float* __restrict__ stats) {
  int t = blockIdx.x * blockDim.x + threadIdx.x;
  if (t < Bsz * Mq * 3) dout[t] = q[t];
  if (t < 512) stats[t] = 0.0f;  // sum0,sq0,sum1,sq1,sum2,sq2
}

// ---------------- ball query: one wave per query point ----------------
__global__ __launch_bounds__(256) void ballquery_kernel(
    const float* __restrict__ p, const float* __restrict__ q,
    int* __restrict__ idx) {
  int wid  = (blockIdx.x * blockDim.x + threadIdx.x) >> 5;  // (b,m)
  int lane = threadIdx.x & 31;
  int b = wid >> 11;  // / Mq
  const float* qp = q + (size_t)wid * 3;
  float qx = qp[0], qy = qp[1], qz = qp[2];
  const float* pb = p + (size_t)b * Npts * 3;
  int* out = idx + (size_t)wid * Kn;

  int cnt = 0, first = 0;
  bool havefirst = false;
  for (int base = 0; base < Npts && cnt < Kn; base += 32) {
    int n = base + lane;
    float px = pb[n * 3 + 0], py = pb[n * 3 + 1], pz = pb[n * 3 + 2];
    float dx = px - qx, dy = py - qy, dz = pz - qz;
    bool cond = (dx * dx + dy * dy + dz * dz) <= R2;
    unsigned mask = __builtin_amdgcn_ballot_w32(cond);
    if (!havefirst && mask) { first = base + __builtin_ctz(mask); havefirst = true; }
    int pos = cnt + __popc(mask & ((1u << lane) - 1u));
    if (cond && pos < Kn) out[pos] = n;
    cnt += __popc(mask);
  }
  if (cnt > Kn) cnt = Kn;
  int pad = havefirst ? first : 0;
  for (int s = cnt + lane; s < Kn; s += 32) out[s] = pad;
}

// ---------------- layer0: gather + GEMM 36(pad)->64 + stats ----------------
__global__ __launch_bounds__(256) void layer0_kernel(
    const float* __restrict__ p, const float* __restrict__ q,
    const float* __restrict__ x, const int* __restrict__ idx,
    const float* __restrict__ W0, const float* __restrict__ bias0,
    float* __restrict__ y0, float* __restrict__ gsum, float* __restrict__ gsq) {
  __shared__ float lsum[C0], lsq[C0];
  int tid = threadIdx.x;
  if (tid < C0) { lsum[tid] = 0.f; lsq[tid] = 0.f; }
  __syncthreads();

  int wave = (blockIdx.x * blockDim.x + tid) >> 5;
  int lane = tid & 31;
  int half = lane >> 4;
  int l    = lane & 15;
  int col  = wave * 16 + l;
  int b    = col >> 16;       // / MK
  int rem  = col & (MK - 1);
  int m    = rem >> 5;        // / Kn
  int n    = idx[col];

  const float* qp = q + ((size_t)(b * Mq + m)) * 3;
  const float* pp = p + ((size_t)b * Npts + n) * 3;
  float ph0 = pp[0] - qp[0], ph1 = pp[1] - qp[1], ph2 = pp[2] - qp[2];
  const float* xb = x + (size_t)b * Cx * Npts + n;  // xb[ci*Npts]

  v8f acc[4] = {};
#pragma unroll
  for (int c = 0; c < 36; c += 4) {
    int cA = c + half * 2;
    float f0, f1;
    // feature cA
    if (cA == 0)       f0 = ph0;
    else if (cA == 1)  f0 = ph1;
    else if (cA == 2)  f0 = ph2;
    else if (cA < 35)  f0 = xb[(size_t)(cA - 3) * Npts];
    else               f0 = 0.f;
    int cB = cA + 1;
    if (cB == 1)       f1 = ph1;
    else if (cB == 2)  f1 = ph2;
    else if (cB < 35)  f1 = xb[(size_t)(cB - 3) * Npts];
    else               f1 = 0.f;
#pragma unroll
    for (int o = 0; o < 4; ++o) {
      int row = o * 16 + l;
      float a0 = (cA < 35) ? W0[row * 35 + cA] : 0.f;
      float a1 = (cB < 35) ? W0[row * 35 + cB] : 0.f;
      acc[o] = wmma4(a0, a1, f0, f1, acc[o]);
    }
  }
#pragma unroll
  for (int o = 0; o < 4; ++o)
#pragma unroll
    for (int v = 0; v < 8; ++v) {
      int ch = o * 16 + v + half * 8;
      float y = acc[o][v] + bias0[ch];
      y0[(size_t)ch * Jtot + col] = y;
      float s1 = xsum16(y);
      float s2 = xsum16(y * y);
      if (l == 0) { atomicAdd(&lsum[ch], s1); atomicAdd(&lsq[ch], s2); }
    }
  __syncthreads();
  if (tid < C0) { atomicAdd(&gsum[tid], lsum[tid]); atomicAdd(&gsq[tid], lsq[tid]); }
}

// ---------------- BN finalize: s = g*rstd, t = be - mean*s ----------------
__global__ void finalize_kernel(const float* __restrict__ sum,
                                const float* __restrict__ sq,
                                const float* __restrict__ g,
                                const float* __restrict__ be,
                                float* __restrict__ sOut, float* __restrict__ tOut,
                                int Cc) {
  int t = blockIdx.x * blockDim.x + threadIdx.x;
  if (t < Cc) {
    float inv = 1.0f / (float)Jtot;
    float mean = sum[t] * inv;
    float var  = sq[t] * inv - mean * mean;
    float rstd = rsqrtf(var + EPSBN);
    float s = g[t] * rstd;
    sOut[t] = s;
    tOut[t] = be[t] - mean * s;
  }
}

// ---------------- mid layers: BN+ReLU(yin) -> GEMM -> [store] + stats ------
template <int CIN, int COUT, bool STORE>
__global__ __launch_bounds__(256) void layer_kernel(
    const float* __restrict__ yin, const float* __restrict__ sC,
    const float* __restrict__ tC, const float* __restrict__ W,
    const float* __restrict__ bias, float* __restrict__ yout,
    float* __restrict__ gsum, float* __restrict__ gsq) {
  __shared__ float lsum[COUT], lsq[COUT];
  int tid = threadIdx.x;
  if (tid < COUT) { lsum[tid] = 0.f; lsq[tid] = 0.f; }
  __syncthreads();

  int wave = (blockIdx.x * blockDim.x + tid) >> 5;
  int lane = tid & 31;
  int half = lane >> 4;
  int l    = lane & 15;
  int col  = wave * 16 + l;
  constexpr int OT = COUT / 16;

  v8f acc[OT] = {};
#pragma unroll
  for (int c = 0; c < CIN; c += 4) {
    int cA = c + half * 2;
    float f0 = fmaxf(yin[(size_t)cA * Jtot + col] * sC[cA] + tC[cA], 0.f);
    float f1 = fmaxf(yin[(size_t)(cA + 1) * Jtot + col] * sC[cA + 1] + tC[cA + 1], 0.f);
#pragma unroll
    for (int o = 0; o < OT; ++o) {
      int row = o * 16 + l;
      float a0 = W[row * CIN + cA];
      float a1 = W[row * CIN + cA + 1];
      acc[o] = wmma4(a0, a1, f0, f1, acc[o]);
    }
  }
#pragma unroll
  for (int o = 0; o < OT; ++o)
#pragma unroll
    for (int v = 0; v < 8; ++v) {
      int ch = o * 16 + v + half * 8;
      float y = acc[o][v] + bias[ch];
      if (STORE) yout[(size_t)ch * Jtot + col] = y;
      float s1 = xsum16(y);
      float s2 = xsum16(y * y);
      if (l == 0) { atomicAdd(&lsum[ch], s1); atomicAdd(&lsq[ch], s2); }
    }
  __syncthreads();
  if (tid < COUT) { atomicAdd(&gsum[tid], lsum[tid]); atomicAdd(&gsq[tid], lsq[tid]); }
}

// ---------------- layer2b: recompute y2, BN+ReLU, max over k, store -------
__global__ __launch_bounds__(256) void layer2b_kernel(
    const float* __restrict__ y1, const float* __restrict__ sIn,
    const float* __restrict__ tIn, const float* __restrict__ W2,
    const float* __restrict__ bias2, const float* __restrict__ s2,
    const float* __restrict__ t2, float* __restrict__ out) {
  int tid  = threadIdx.x;
  int wave = (blockIdx.x * blockDim.x + tid) >> 5;  // one wave per (b,m)
  int lane = tid & 31;
  int half = lane >> 4;
  int l    = lane & 15;
  int b    = wave >> 11;        // / Mq
  int m    = wave & (Mq - 1);

  float mx[8][8];
#pragma unroll
  for (int t = 0; t < 2; ++t) {            // k halves: 0..15 / 16..31
    int col = wave * 32 + t * 16 + l;
    v8f acc[8] = {};
#pragma unroll
    for (int c = 0; c < 64; c += 4) {
      int cA = c + half * 2;
      float f0 = fmaxf(y1[(size_t)cA * Jtot + col] * sIn[cA] + tIn[cA], 0.f);
      float f1 = fmaxf(y1[(size_t)(cA + 1) * Jtot + col] * sIn[cA + 1] + tIn[cA + 1], 0.f);
#pragma unroll
      for (int o = 0; o < 8; ++o) {
        int row = o * 16 + l;
        acc[o] = wmma4(W2[row * 64 + cA], W2[row * 64 + cA + 1], f0, f1, acc[o]);
      }
    }
#pragma unroll
    for (int o = 0; o < 8; ++o)
#pragma unroll
      for (int v = 0; v < 8; ++v) {
        int ch = o * 16 + v + half * 8;
        float yv = fmaxf((acc[o][v] + bias2[ch]) * s2[ch] + t2[ch], 0.f);
        mx[o][v] = (t == 0) ? yv : fmaxf(mx[o][v], yv);
      }
  }
#pragma unroll
  for (int o = 0; o < 8; ++o)
#pragma unroll
    for (int v = 0; v < 8; ++v) {
      float vm = xmax16(mx[o][v]);
      if (l == 0) {
        int ch = o * 16 + v + half * 8;
        out[((size_t)b * C2 + ch) * Mq + m] = vm;
      }
    }
}

// ---------------- host launch ----------------
extern "C" void kernel_launch(void* const* d_in, const int* in_sizes, int n_in,
                              void* d_out, int out_size, void* d_ws, size_t ws_size,
                              hipStream_t stream) {
  const float* p  = (const float*)d_in[0];
  const float* q  = (const float*)d_in[1];
  const float* x  = (const float*)d_in[2];
  const float* w0 = (const float*)d_in[3];
  const float* b0 = (const float*)d_in[4];
  const float* g0 = (const float*)d_in[5];
  const float* be0= (const float*)d_in[6];
  const float* w1 = (const float*)d_in[7];
  const float* b1 = (const float*)d_in[8];
  const float* g1 = (const float*)d_in[9];
  const float* be1= (const float*)d_in[10];
  const float* w2 = (const float*)d_in[11];
  const float* b2 = (const float*)d_in[12];
  const float* g2 = (const float*)d_in[13];
  const float* be2= (const float*)d_in[14];

  char* ws = (char*)d_ws;
  int*   idx = (int*)ws;                                   // Jtot ints (1 MB)
  float* y0  = (float*)(ws + (size_t)Jtot * 4);            // 64*J floats (67 MB)
  float* y1  = (float*)(ws + (size_t)Jtot * 4 + (size_t)C0 * Jtot * 4);
  float* st  = (float*)(ws + (size_t)Jtot * 4 + (size_t)(C0 + C1) * Jtot * 4);
  float* sum0 = st +   0, * sq0 = st +  64;
  float* sum1 = st + 128, * sq1 = st + 192;
  float* sum2 = st + 256, * sq2 = st + 384;
  float* s0 = st + 512, * t0 = st + 576;
  float* s1 = st + 640, * t1 = st + 704;
  float* s2 = st + 768, * t2 = st + 896;

  float* outq = (float*)d_out;
  float* outf = outq + (size_t)Bsz * Mq * 3;

  init_kernel<<<96, 256, 0, stream>>>(q, outq, st);                       // 24576 threads
  ballquery_kernel<<<(Bsz * Mq) / 8, 256, 0, stream>>>(p, q, idx);        // 8192 waves
  layer0_kernel<<<(Jtot / 16) / 8, 256, 0, stream>>>(p, q, x, idx, w0, b0,
                                                     y0, sum0, sq0);
  finalize_kernel<<<1, 64, 0, stream>>>(sum0, sq0, g0, be0, s0, t0, C0);
  layer_kernel<C0, C1, true><<<(Jtot / 16) / 8, 256, 0, stream>>>(
      y0, s0, t0, w1, b1, y1, sum1, sq1);
  finalize_kernel<<<1, 64, 0, stream>>>(sum1, sq1, g1, be1, s1, t1, C1);
  layer_kernel<C1, C2, false><<<(Jtot / 16) / 8, 256, 0, stream>>>(
      y1, s1, t1, w2, b2, (float*)nullptr, sum2, sq2);
  finalize_kernel<<<1, 128, 0, stream>>>(sum2, sq2, g2, be2, s2, t2, C2);
  layer2b_kernel<<<(Bsz * Mq) / 8, 256, 0, stream>>>(y1, s1, t1, w2, b2, s2,
                                                     t2, outf);
}